// HypergraphNet_6476810682471
// MI455X (gfx1250) — compile-verified
//
#include <hip/hip_runtime.h>
#include <hip/hip_bf16.h>

typedef __attribute__((ext_vector_type(2))) float v2f;
typedef __attribute__((ext_vector_type(8))) float v8f;

#define N_NODES      500000
#define N_INCIDENCE  2000000
#define N_HYPEREDGES 100000
#define N_GRAPHS     512
#define CH           64

// ---------------------------------------------------------------------------
// Generic zero-fill
// ---------------------------------------------------------------------------
__global__ void zero_kernel(float* __restrict__ p, long long n) {
    long long i = (long long)blockIdx.x * blockDim.x + threadIdx.x;
    if (i < n) p[i] = 0.0f;
}

// ---------------------------------------------------------------------------
// Degree accumulation: D[row[i]] += 1, B[col[i]] += 1
// ---------------------------------------------------------------------------
__global__ void degree_kernel(const int* __restrict__ row, const int* __restrict__ col,
                              float* __restrict__ D, float* __restrict__ B, int nnz) {
    int i = blockIdx.x * blockDim.x + threadIdx.x;
    if (i < nnz) {
        atomicAdd(&D[row[i]], 1.0f);
        atomicAdd(&B[col[i]], 1.0f);
    }
}

// In-place: p[i] = p[i] > 0 ? 1/p[i] : 0
__global__ void invert_kernel(float* __restrict__ p, int n) {
    int i = blockIdx.x * blockDim.x + threadIdx.x;
    if (i < n) {
        float v = p[i];
        p[i] = (v > 0.0f) ? (1.0f / v) : 0.0f;
    }
}

// ---------------------------------------------------------------------------
// WMMA f32 GEMM: Y[nrows,64] = X[nrows,64] @ W[64,64] + bias[64]
// One wave32 per 16x16 output tile. Uses V_WMMA_F32_16X16X4_F32, K=64 in 16 steps.
//
// A (16x4 f32) layout per ISA: v0 = {lanes0-15: K=0, lanes16-31: K=2},
//                              v1 = {lanes0-15: K=1, lanes16-31: K=3}, M = lane&15.
// B (4x16 f32): one K-row striped across 16 lanes per VGPR half, N = lane&15.
// C/D (16x16 f32): VGPR j holds M=j (lanes0-15) and M=8+j (lanes16-31), N = lane&15.
// ---------------------------------------------------------------------------
__global__ void gemm_wmma_f32(const float* __restrict__ X, const float* __restrict__ W,
                              const float* __restrict__ bias, float* __restrict__ Y,
                              int nrows) {
    const int wave  = (blockIdx.x * blockDim.x + threadIdx.x) >> 5;
    const int lane  = threadIdx.x & 31;
    const int tiles_n = CH / 16;                 // 4
    const int tile_m  = wave / tiles_n;
    const int tile_n  = wave % tiles_n;
    const int m0 = tile_m * 16;
    if (m0 >= nrows) return;                     // uniform per wave -> EXEC stays all-1
    const int n0   = tile_n * 16;
    const int half = lane >> 4;                  // 0: K even pair low, 1: K pair high
    const int r    = lane & 15;

    const float* __restrict__ xrow = X + (size_t)(m0 + r) * CH;

    v8f c = {};
#pragma unroll
    for (int k = 0; k < CH; k += 4) {
        const int ka = k + 2 * half;
        v2f a = *(const v2f*)(xrow + ka);        // A[M=r][ka], A[M=r][ka+1]
        v2f b;
        b.x = W[(size_t)ka * CH + n0 + r];       // B[K=ka  ][N=r]
        b.y = W[(size_t)(ka + 1) * CH + n0 + r]; // B[K=ka+1][N=r]
        c = __builtin_amdgcn_wmma_f32_16x16x4_f32(false, a, false, b,
                                                  (short)0, c, false, false);
    }

    const float bv = bias[n0 + r];
#pragma unroll
    for (int j = 0; j < 8; ++j) {
        Y[(size_t)(m0 + half * 8 + j) * CH + n0 + r] = c[j] + bv;
    }
}

// ---------------------------------------------------------------------------
// Sparse propagate: dst[sidx[i]][:] += src[gidx[i]][:]
// 16 lanes per incidence, float4 per lane (global_load_b128 + 4x atomic f32).
// ---------------------------------------------------------------------------
__global__ void gather_scatter_kernel(const float* __restrict__ src, float* __restrict__ dst,
                                      const int* __restrict__ gidx, const int* __restrict__ sidx,
                                      int nnz) {
    long long t  = (long long)blockIdx.x * blockDim.x + threadIdx.x;
    int inc = (int)(t >> 4);
    int q   = (int)(t & 15);
    if (inc >= nnz) return;
    const int g = gidx[inc];
    const int s = sidx[inc];
    const float4 v = *(const float4*)(src + (size_t)g * CH + q * 4);
    float* d = dst + (size_t)s * CH + q * 4;
    atomicAdd(d + 0, v.x);
    atomicAdd(d + 1, v.y);
    atomicAdd(d + 2, v.z);
    atomicAdd(d + 3, v.w);
}

// rows scaled by per-row factor, optional ReLU
__global__ void scale_rows_kernel(float* __restrict__ p, const float* __restrict__ scal,
                                  int nrows, int relu) {
    long long t = (long long)blockIdx.x * blockDim.x + threadIdx.x;
    int rw = (int)(t >> 4);
    int q  = (int)(t & 15);
    if (rw >= nrows) return;
    const float s = scal[rw];
    float4* ptr = (float4*)(p + (size_t)rw * CH + q * 4);
    float4 v = *ptr;
    v.x *= s; v.y *= s; v.z *= s; v.w *= s;
    if (relu) {
        v.x = fmaxf(v.x, 0.0f); v.y = fmaxf(v.y, 0.0f);
        v.z = fmaxf(v.z, 0.0f); v.w = fmaxf(v.w, 0.0f);
    }
    *ptr = v;
}

// ---------------------------------------------------------------------------
// Global mean pool + FC head
// ---------------------------------------------------------------------------
__global__ void pool_sum_kernel(const float* __restrict__ h, const int* __restrict__ batch,
                                float* __restrict__ pooled, int nrows) {
    long long t = (long long)blockIdx.x * blockDim.x + threadIdx.x;
    int rw = (int)(t >> 4);
    int q  = (int)(t & 15);
    if (rw >= nrows) return;
    const int g = batch[rw];
    const float4 v = *(const float4*)(h + (size_t)rw * CH + q * 4);
    float* d = pooled + (size_t)g * CH + q * 4;
    atomicAdd(d + 0, v.x);
    atomicAdd(d + 1, v.y);
    atomicAdd(d + 2, v.z);
    atomicAdd(d + 3, v.w);
}

__global__ void pool_count_kernel(const int* __restrict__ batch, float* __restrict__ counts,
                                  int nrows) {
    int i = blockIdx.x * blockDim.x + threadIdx.x;
    if (i < nrows) atomicAdd(&counts[batch[i]], 1.0f);
}

__global__ void final_fc_kernel(const float* __restrict__ pooled, const float* __restrict__ counts,
                                const float* __restrict__ Wfc, const float* __restrict__ bfc,
                                float* __restrict__ out) {
    int g = blockIdx.x * blockDim.x + threadIdx.x;
    if (g >= N_GRAPHS) return;
    const float inv = 1.0f / fmaxf(counts[g], 1.0f);
    float acc = 0.0f;
#pragma unroll 8
    for (int k = 0; k < CH; ++k) acc += pooled[(size_t)g * CH + k] * Wfc[k];
    out[g] = acc * inv + bfc[0];
}

// ---------------------------------------------------------------------------
// Host orchestration
// ---------------------------------------------------------------------------
static inline void launch_zero(float* p, long long n, hipStream_t s) {
    int blocks = (int)((n + 255) / 256);
    zero_kernel<<<blocks, 256, 0, s>>>(p, n);
}

extern "C" void kernel_launch(void* const* d_in, const int* in_sizes, int n_in,
                              void* d_out, int out_size, void* d_ws, size_t ws_size,
                              hipStream_t stream) {
    const float* x     = (const float*)d_in[0];
    const int*   ei    = (const int*)d_in[1];
    const int*   batch = (const int*)d_in[2];
    const float* W1    = (const float*)d_in[3];
    const float* b1    = (const float*)d_in[4];
    const float* W2    = (const float*)d_in[5];
    const float* b2    = (const float*)d_in[6];
    const float* Wfc   = (const float*)d_in[7];
    const float* bfc   = (const float*)d_in[8];
    float* out = (float*)d_out;

    const int* row = ei;                // [nnz]
    const int* col = ei + N_INCIDENCE;  // [nnz]

    float* ws = (float*)d_ws;
    float* xt     = ws;                              // 32,000,000 f32
    float* hb     = ws + 32000000LL;                 // 32,000,000 f32
    float* mb     = ws + 64000000LL;                 //  6,400,000 f32
    float* Dinv   = ws + 70400000LL;                 //    500,000 f32
    float* Binv   = ws + 70900000LL;                 //    100,000 f32
    float* pooled = ws + 71000000LL;                 //     32,768 f32
    float* counts = ws + 71032768LL;                 //        512 f32

    const long long node_elems = (long long)N_NODES * CH;
    const long long edge_elems = (long long)N_HYPEREDGES * CH;

    // --- degrees (shared by both layers) ---
    launch_zero(Dinv, N_NODES, stream);
    launch_zero(Binv, N_HYPEREDGES, stream);
    degree_kernel<<<(N_INCIDENCE + 255) / 256, 256, 0, stream>>>(row, col, Dinv, Binv, N_INCIDENCE);
    invert_kernel<<<(N_NODES + 255) / 256, 256, 0, stream>>>(Dinv, N_NODES);
    invert_kernel<<<(N_HYPEREDGES + 255) / 256, 256, 0, stream>>>(Binv, N_HYPEREDGES);

    const int gemm_waves  = (N_NODES / 16) * (CH / 16);      // 125,000
    const int gemm_blocks = gemm_waves / 8;                  // 8 waves per 256-thr block
    const int inc_blocks  = (int)(((long long)N_INCIDENCE * 16 + 255) / 256);
    const int node_blocks = (int)(((long long)N_NODES * 16 + 255) / 256);
    const int edge_blocks = (int)(((long long)N_HYPEREDGES * 16 + 255) / 256);

    // ================= Layer 1 =================
    gemm_wmma_f32<<<gemm_blocks, 256, 0, stream>>>(x, W1, b1, xt, N_NODES);

    launch_zero(mb, edge_elems, stream);
    gather_scatter_kernel<<<inc_blocks, 256, 0, stream>>>(xt, mb, row, col, N_INCIDENCE);
    scale_rows_kernel<<<edge_blocks, 256, 0, stream>>>(mb, Binv, N_HYPEREDGES, 0);

    launch_zero(hb, node_elems, stream);
    gather_scatter_kernel<<<inc_blocks, 256, 0, stream>>>(mb, hb, col, row, N_INCIDENCE);
    scale_rows_kernel<<<node_blocks, 256, 0, stream>>>(hb, Dinv, N_NODES, 1);  // * Dinv + ReLU

    // ================= Layer 2 =================
    gemm_wmma_f32<<<gemm_blocks, 256, 0, stream>>>(hb, W2, b2, xt, N_NODES);

    launch_zero(mb, edge_elems, stream);
    gather_scatter_kernel<<<inc_blocks, 256, 0, stream>>>(xt, mb, row, col, N_INCIDENCE);
    scale_rows_kernel<<<edge_blocks, 256, 0, stream>>>(mb, Binv, N_HYPEREDGES, 0);

    launch_zero(hb, node_elems, stream);
    gather_scatter_kernel<<<inc_blocks, 256, 0, stream>>>(mb, hb, col, row, N_INCIDENCE);
    scale_rows_kernel<<<node_blocks, 256, 0, stream>>>(hb, Dinv, N_NODES, 1);

    // ================= Pool + FC =================
    launch_zero(pooled, (long long)N_GRAPHS * CH, stream);
    launch_zero(counts, N_GRAPHS, stream);
    pool_sum_kernel<<<node_blocks, 256, 0, stream>>>(hb, batch, pooled, N_NODES);
    pool_count_kernel<<<(N_NODES + 255) / 256, 256, 0, stream>>>(batch, counts, N_NODES);
    final_fc_kernel<<<(N_GRAPHS + 255) / 256, 256, 0, stream>>>(pooled, counts, Wfc, bfc, out);
}